// DGCN_68702296866875
// MI455X (gfx1250) — compile-verified
//
#include <hip/hip_runtime.h>
#include <hip/hip_bf16.h>

// ---------------- problem constants ----------------
#define ROI   268
#define NB    64
#define NN    (ROI * NB)   // 17152 nodes
#define FDIM  268          // feature dim (== ROI by coincidence)
#define CDIM  32
#define DEG   32
#define KNN   32
#define PW    144          // padded width of projection matrix P (130 used)
#define ZW    332          // readout input width: 32 + 32 + 268
#define SLOPE 0.33f
#define BNEPS 1e-5f

typedef __attribute__((ext_vector_type(2))) float v2f;
typedef __attribute__((ext_vector_type(8))) float v8f;

__device__ __forceinline__ float leaky(float v) { return v >= 0.f ? v : SLOPE * v; }

// D(16x16,f32) += A(16x4,f32) * B(4x16,f32)   -- CDNA5 V_WMMA_F32_16X16X4_F32
__device__ __forceinline__ v8f wmma4(v2f a, v2f b, v8f c) {
  return __builtin_amdgcn_wmma_f32_16x16x4_f32(false, a, false, b, (short)0, c,
                                               false, false);
}

// Async global -> LDS copy, 16 bytes per lane (GLOBAL_LOAD_ASYNC_TO_LDS_B128).
// ldsOff = wave-relative LDS byte offset (low 32 bits of a generic __shared__
// pointer); gaddr = 64-bit global address. Tracked by ASYNCcnt.
__device__ __forceinline__ void async_ld_b128(unsigned ldsOff, const void* gaddr) {
  asm volatile("global_load_async_to_lds_b128 %0, %1, off"
               :: "v"(ldsOff), "v"(gaddr) : "memory");
}
__device__ __forceinline__ void wait_async0() {
  asm volatile("s_wait_asynccnt 0" ::: "memory");
}

// ---------------------------------------------------------------------------
// Kernel 1: pack Wcat (268 x 144):
//  cols 0-31  = W1a[:F]   cols 32-63 = W1a[F:]
//  cols 64-95 = W2a[:F]   cols 96-127= W2a[F:]
//  col 128    = W3[:F]    col 129    = W3[F:]   cols 130-143 = 0
// ---------------------------------------------------------------------------
__global__ void pack_wcat_kernel(const float* __restrict__ W1a,
                                 const float* __restrict__ W2a,
                                 const float* __restrict__ W3,
                                 float* __restrict__ Wcat) {
  int idx = blockIdx.x * blockDim.x + threadIdx.x;
  if (idx >= FDIM * PW) return;
  int f = idx / PW, c = idx % PW;
  float v = 0.f;
  if (c < 32)        v = W1a[f * CDIM + c];
  else if (c < 64)   v = W1a[(FDIM + f) * CDIM + (c - 32)];
  else if (c < 96)   v = W2a[f * CDIM + (c - 64)];
  else if (c < 128)  v = W2a[(FDIM + f) * CDIM + (c - 96)];
  else if (c == 128) v = W3[f];
  else if (c == 129) v = W3[FDIM + f];
  Wcat[idx] = v;
}

// ---------------------------------------------------------------------------
// Kernel 2: P = X(NN x 268) @ Wcat(268 x 144) via f32 WMMA 16x16x4.
// Block = 9 waves; each wave owns one 16-wide column tile, block owns 16 rows.
// X tile staged in LDS with async b128 copies; Wcat stays L2-resident.
// ---------------------------------------------------------------------------
__global__ __launch_bounds__(288) void proj_gemm_kernel(
    const float* __restrict__ X, const float* __restrict__ Wcat,
    float* __restrict__ P) {
  __shared__ __align__(16) float Xs[16 * FDIM];
  const int rowBase = blockIdx.x * 16;
  // 16 rows x 268 f32 = 1072 16-byte vectors; rows are 1072 B (16B multiple).
  for (int t = threadIdx.x; t < 16 * (FDIM / 4); t += 288) {
    int r = t / (FDIM / 4), c4 = (t % (FDIM / 4)) * 4;
    async_ld_b128((unsigned)(size_t)&Xs[r * FDIM + c4],
                  &X[(size_t)(rowBase + r) * FDIM + c4]);
  }
  wait_async0();
  __syncthreads();

  const int wave = threadIdx.x >> 5;      // 0..8 -> column tile
  const int lane = threadIdx.x & 31;
  const int colBase = wave * 16;
  const int half = lane >> 4;             // lanes 16-31 carry K+2,K+3
  const int l15  = lane & 15;

  v8f acc = {};
  for (int k = 0; k < FDIM / 4; ++k) {
    const int k0 = k * 4 + half * 2;
    v2f a, b;
    a.x = Xs[l15 * FDIM + k0];
    a.y = Xs[l15 * FDIM + k0 + 1];
    b.x = Wcat[(size_t)k0 * PW + colBase + l15];
    b.y = Wcat[(size_t)(k0 + 1) * PW + colBase + l15];
    // unconditional: OOB prefetch lands in mapped workspace, harmless
    __builtin_prefetch(&Wcat[(size_t)(k0 + 32) * PW + colBase], 0, 1);
    acc = wmma4(a, b, acc);
  }
  // C/D layout: VGPR v: lanes0-15 -> M=v, lanes16-31 -> M=v+8, N = lane&15
  for (int v = 0; v < 8; ++v) {
    int r = rowBase + v + half * 8;
    P[(size_t)r * PW + colBase + l15] = acc[v];
  }
}

// ---------------------------------------------------------------------------
// Kernel 3: kNN (k=32, includes self) within each 268-node graph.
// One block per target node: distances in LDS, 32 rounds of argmin
// (tie-break on lower index, matching top_k).
// ---------------------------------------------------------------------------
__global__ __launch_bounds__(256) void knn_kernel(const float* __restrict__ X,
                                                  int* __restrict__ knnIdx) {
  __shared__ float xi[FDIM];
  __shared__ float dist[ROI];
  __shared__ float sVal[256];
  __shared__ int   sIdx[256];
  const int i = blockIdx.x;
  const int base = (i / ROI) * ROI;
  for (int t = threadIdx.x; t < FDIM; t += 256) xi[t] = X[(size_t)i * FDIM + t];
  __syncthreads();
  for (int j = threadIdx.x; j < ROI; j += 256) {
    const float* xj = X + (size_t)(base + j) * FDIM;
    float s = 0.f;
    for (int f = 0; f < FDIM; ++f) { float d = xj[f] - xi[f]; s += d * d; }
    dist[j] = s;
  }
  __syncthreads();
  for (int k = 0; k < KNN; ++k) {
    float bv = 3.0e38f; int bi = ROI;
    for (int j = threadIdx.x; j < ROI; j += 256) {
      float v = dist[j];
      if (v < bv) { bv = v; bi = j; }
    }
    sVal[threadIdx.x] = bv; sIdx[threadIdx.x] = bi;
    __syncthreads();
    for (int s = 128; s > 0; s >>= 1) {
      if (threadIdx.x < s) {
        float ov = sVal[threadIdx.x + s]; int oi = sIdx[threadIdx.x + s];
        if (ov < sVal[threadIdx.x] ||
            (ov == sVal[threadIdx.x] && oi < sIdx[threadIdx.x])) {
          sVal[threadIdx.x] = ov; sIdx[threadIdx.x] = oi;
        }
      }
      __syncthreads();
    }
    if (threadIdx.x == 0) {
      knnIdx[(size_t)i * KNN + k] = base + sIdx[0];
      dist[sIdx[0]] = 3.0e38f;
    }
    __syncthreads();
  }
}

// ---------------------------------------------------------------------------
// Kernel 4: fused EdgeConv: per node i (one wave), 32 neighbors:
//   h[l][c] = leaky(Pi[piOff+c] + Pj[pdOff+c] - Pi[pdOff+c] + ba[c])
//   M = h(32x32) @ W(32x32)  (WMMA, 2x2 tiles of 16x16, K chained 8x4)
//   outA[i][c] = max_l M[l][c] + bb[c]
// Optionally also the scalar gcn3 max (P cols 128/129).
// W staged into LDS with async b128 copies (4 KB, one b128 per thread).
// ---------------------------------------------------------------------------
__global__ __launch_bounds__(256) void edge_conv_kernel(
    const float* __restrict__ P, const int* __restrict__ gsrc,
    int piOff, int pdOff, const float* __restrict__ W,
    const float* __restrict__ ba, const float* __restrict__ bb,
    float* __restrict__ outA, int doScalar, const float* __restrict__ b3,
    float* __restrict__ a3) {
  __shared__ __align__(16) float Ws[32 * 32];
  __shared__ float hs[8][32][33];
  const int wave = threadIdx.x >> 5;
  const int lane = threadIdx.x & 31;
  async_ld_b128((unsigned)(size_t)&Ws[threadIdx.x * 4], &W[threadIdx.x * 4]);

  const int i = blockIdx.x * 8 + wave;
  const int j = gsrc[(size_t)i * 32 + lane];
  const float* Pi = P + (size_t)i * PW;
  const float* Pj = P + (size_t)j * PW;
  for (int c = 0; c < 32; ++c) {
    float h = Pi[piOff + c] + Pj[pdOff + c] - Pi[pdOff + c] + ba[c];
    hs[wave][lane][c] = leaky(h);
  }
  if (doScalar) {
    float s = Pi[128] + Pj[129] - Pi[129] + b3[0];
    for (int d = 16; d > 0; d >>= 1) s = fmaxf(s, __shfl_xor(s, d, 32));
    if (lane == 0) a3[i] = s;
  }
  wait_async0();
  __syncthreads();

  const int half = lane >> 4, l15 = lane & 15;
  v8f acc[2][2];
  for (int rt = 0; rt < 2; ++rt)
    for (int ct = 0; ct < 2; ++ct) {
      v8f c = {};
      for (int k = 0; k < 8; ++k) {
        const int k0 = k * 4 + half * 2;
        v2f a, b;
        a.x = hs[wave][rt * 16 + l15][k0];
        a.y = hs[wave][rt * 16 + l15][k0 + 1];
        b.x = Ws[k0 * 32 + ct * 16 + l15];
        b.y = Ws[(k0 + 1) * 32 + ct * 16 + l15];
        c = wmma4(a, b, c);
      }
      acc[rt][ct] = c;
    }
  // column-max over the 32 edge rows, then +bias, store
  for (int ct = 0; ct < 2; ++ct) {
    float m = -3.0e38f;
    for (int v = 0; v < 8; ++v)
      m = fmaxf(m, fmaxf(acc[0][ct][v], acc[1][ct][v]));
    m = fmaxf(m, __shfl_xor(m, 16, 32));
    int col = ct * 16 + l15;
    if (lane < 16) outA[(size_t)i * 32 + col] = m + bb[col];
  }
}

// ---------------------------------------------------------------------------
// Kernel 5: batch-norm statistics (training mode, biased var) -> scale/shift.
// One block per channel. ss[c] = g/sqrt(var+eps), ss[C+c] = be - mu*scale.
// ---------------------------------------------------------------------------
__global__ void bn_stats_kernel(const float* __restrict__ a, int n, int stride,
                                const float* __restrict__ g,
                                const float* __restrict__ be,
                                float* __restrict__ ss, int Ccount) {
  __shared__ float s1[256], s2[256];
  const int c = blockIdx.x;
  float s = 0.f, q = 0.f;
  for (int i = threadIdx.x; i < n; i += blockDim.x) {
    float v = a[(size_t)i * stride + c];
    s += v; q += v * v;
  }
  s1[threadIdx.x] = s; s2[threadIdx.x] = q;
  __syncthreads();
  for (int st = blockDim.x >> 1; st > 0; st >>= 1) {
    if ((int)threadIdx.x < st) {
      s1[threadIdx.x] += s1[threadIdx.x + st];
      s2[threadIdx.x] += s2[threadIdx.x + st];
    }
    __syncthreads();
  }
  if (threadIdx.x == 0) {
    float mu = s1[0] / n;
    float var = s2[0] / n - mu * mu;
    float inv = rsqrtf(var + BNEPS);
    float sc = g[c] * inv;
    ss[c] = sc;
    ss[Ccount + c] = be[c] - mu * sc;
  }
}

// ---------------------------------------------------------------------------
// Kernel 6: per-graph mean pooling of leaky(bn(a)) into z columns [zoff, zoff+C)
// One 64-thread block per (graph, channel).
// ---------------------------------------------------------------------------
__global__ __launch_bounds__(64) void pool_kernel(const float* __restrict__ a,
                                                  const float* __restrict__ ss,
                                                  int Ccount,
                                                  float* __restrict__ z,
                                                  int zoff) {
  __shared__ float red[64];
  const int b = blockIdx.x / Ccount, c = blockIdx.x % Ccount;
  const float sc = ss[c], sh = ss[Ccount + c];
  float s = 0.f;
  for (int r = threadIdx.x; r < ROI; r += 64)
    s += leaky(a[(size_t)(b * ROI + r) * Ccount + c] * sc + sh);
  red[threadIdx.x] = s;
  __syncthreads();
  for (int st = 32; st > 0; st >>= 1) {
    if ((int)threadIdx.x < st) red[threadIdx.x] += red[threadIdx.x + st];
    __syncthreads();
  }
  if (threadIdx.x == 0) z[(size_t)b * ZW + zoff + c] = red[0] / (float)ROI;
}

// Kernel 7: x3 = leaky(bn(a3)) into z columns 64..331
__global__ void x3_kernel(const float* __restrict__ a3,
                          const float* __restrict__ ss3,
                          float* __restrict__ z) {
  int i = blockIdx.x * 256 + threadIdx.x;
  if (i >= NN) return;
  float v = leaky(a3[i] * ss3[0] + ss3[1]);
  int b = i / ROI, r = i % ROI;
  z[(size_t)b * ZW + 2 * CDIM + r] = v;
}

// ---------------------------------------------------------------------------
// Kernel 8: small FC: out[b][o] = sum_k act(in[b][k]) * W[k][o] + bias[o]
// act = identity if ssIn==null else leaky(bn) via precomputed scale/shift.
// ---------------------------------------------------------------------------
__global__ void fc_kernel(const float* __restrict__ in, int inC,
                          const float* __restrict__ ssIn,
                          const float* __restrict__ W,
                          const float* __restrict__ bias,
                          float* __restrict__ out, int outC) {
  int idx = blockIdx.x * blockDim.x + threadIdx.x;
  if (idx >= NB * outC) return;
  int b = idx / outC, o = idx % outC;
  float s = bias[o];
  for (int k = 0; k < inC; ++k) {
    float v = in[(size_t)b * inC + k];
    if (ssIn) v = leaky(v * ssIn[k] + ssIn[inC + k]);
    s += v * W[(size_t)k * outC + o];
  }
  out[(size_t)b * outC + o] = s;
}

// ---------------------------------------------------------------------------
extern "C" void kernel_launch(void* const* d_in, const int* in_sizes, int n_in,
                              void* d_out, int out_size, void* d_ws,
                              size_t ws_size, hipStream_t stream) {
  (void)in_sizes; (void)n_in; (void)out_size; (void)ws_size;
  const float* x   = (const float*)d_in[0];
  const int* eidx  = (const int*)d_in[1];     // [2, E]: row0=src, row1=tgt
  const float* W1a = (const float*)d_in[3];
  const float* b1a = (const float*)d_in[4];
  const float* W1b = (const float*)d_in[5];
  const float* b1b = (const float*)d_in[6];
  const float* g1  = (const float*)d_in[7];
  const float* be1 = (const float*)d_in[8];
  const float* W2a = (const float*)d_in[9];
  const float* b2a = (const float*)d_in[10];
  const float* W2b = (const float*)d_in[11];
  const float* b2b = (const float*)d_in[12];
  const float* g2  = (const float*)d_in[13];
  const float* be2 = (const float*)d_in[14];
  const float* W3  = (const float*)d_in[15];
  const float* b3  = (const float*)d_in[16];
  const float* g3  = (const float*)d_in[17];
  const float* be3 = (const float*)d_in[18];
  const float* Wl1 = (const float*)d_in[19];
  const float* bl1 = (const float*)d_in[20];
  const float* g4  = (const float*)d_in[21];
  const float* be4 = (const float*)d_in[22];
  const float* Wl2 = (const float*)d_in[23];
  const float* bl2 = (const float*)d_in[24];
  const float* g5  = (const float*)d_in[25];
  const float* be5 = (const float*)d_in[26];
  const float* Wl3 = (const float*)d_in[27];
  const float* bl3 = (const float*)d_in[28];
  float* out = (float*)d_out;

  // workspace carve-up (~17 MB, all fully written before read)
  float* wsf  = (float*)d_ws;
  float* Wcat = wsf;                 wsf += FDIM * PW;
  float* P    = wsf;                 wsf += (size_t)NN * PW;
  float* a1   = wsf;                 wsf += (size_t)NN * CDIM;
  float* m2a  = wsf;                 wsf += (size_t)NN * CDIM;
  float* a3   = wsf;                 wsf += NN;
  int*   knnI = (int*)wsf;           wsf += (size_t)NN * KNN;
  float* ss1  = wsf;                 wsf += 2 * CDIM;
  float* ss2  = wsf;                 wsf += 2 * CDIM;
  float* ss3  = wsf;                 wsf += 2;
  float* z    = wsf;                 wsf += (size_t)NB * ZW;
  float* u1   = wsf;                 wsf += (size_t)NB * 256;
  float* ss4  = wsf;                 wsf += 512;
  float* u2   = wsf;                 wsf += (size_t)NB * 128;
  float* ss5  = wsf;                 wsf += 256;

  const int* src = eidx;  // first E=NN*32 entries; tgt(e)=e/32 implicitly

  pack_wcat_kernel<<<(FDIM * PW + 255) / 256, 256, 0, stream>>>(W1a, W2a, W3, Wcat);
  proj_gemm_kernel<<<NN / 16, 288, 0, stream>>>(x, Wcat, P);
  knn_kernel<<<NN, 256, 0, stream>>>(x, knnI);

  // gcn1 (+ gcn3 scalar): src edges, P cols [0..31]=P1i, [32..63]=P1d
  edge_conv_kernel<<<NN / 8, 256, 0, stream>>>(P, src, 0, 32, W1b, b1a, b1b,
                                               a1, 1, b3, a3);
  // gcn2: knn neighbors, P cols [64..95]=P2i, [96..127]=P2d
  edge_conv_kernel<<<NN / 8, 256, 0, stream>>>(P, knnI, 64, 96, W2b, b2a, b2b,
                                               m2a, 0, nullptr, nullptr);

  bn_stats_kernel<<<CDIM, 256, 0, stream>>>(a1, NN, CDIM, g1, be1, ss1, CDIM);
  bn_stats_kernel<<<CDIM, 256, 0, stream>>>(m2a, NN, CDIM, g2, be2, ss2, CDIM);
  bn_stats_kernel<<<1, 256, 0, stream>>>(a3, NN, 1, g3, be3, ss3, 1);

  pool_kernel<<<NB * CDIM, 64, 0, stream>>>(a1, ss1, CDIM, z, 0);
  pool_kernel<<<NB * CDIM, 64, 0, stream>>>(m2a, ss2, CDIM, z, CDIM);
  x3_kernel<<<NN / 256, 256, 0, stream>>>(a3, ss3, z);

  fc_kernel<<<(NB * 256) / 256, 256, 0, stream>>>(z, ZW, nullptr, Wl1, bl1, u1, 256);
  bn_stats_kernel<<<256, 64, 0, stream>>>(u1, NB, 256, g4, be4, ss4, 256);
  fc_kernel<<<(NB * 128) / 256, 256, 0, stream>>>(u1, 256, ss4, Wl2, bl2, u2, 128);
  bn_stats_kernel<<<128, 64, 0, stream>>>(u2, NB, 128, g5, be5, ss5, 128);
  fc_kernel<<<1, 64, 0, stream>>>(u2, 128, ss5, Wl3, bl3, out, 1);
}